// MarginalFlowBank_21208548507743
// MI455X (gfx1250) — compile-verified
//
#include <hip/hip_runtime.h>
#include <math.h>

// Rational-quadratic spline (neural spline flow) per-dim log-prob.
// x: (N, D=32) row-major; per-dim K=8 bins, linear tails outside [-10, 10].
//
// Memory-bound streaming kernel: 268 MB traffic -> ~11.5 us at 23.3 TB/s HBM.
// Per-element VALU budget at that rate is ~28 lane-ops, so the inner loop is
// written for minimum VALU: fast log (v_log_f32), raw v_rcp_f32, binary bin
// search (3 cmp + 4 sel + v_add3 -> byte offset), fma-folded spline math, and
// a 24 B LDS gather (ds_load_b128 + ds_load_b64) per element. VOPD dual-issue
// and the TRANS pipe cover the remainder.

#define KBINS 8
#define DIMS 32
#define BOUND 10.0f
#define MINW 1e-3f
#define MINH 1e-3f
#define MIND 1e-3f
#define HALF_LOG_2PI 0.91893853320467274178f
#define ROWB 288  // bytes per dim row: 8 records * 32 B + 32 B pad (bank spread)

// record (32 B, per dim*bin): {-cw*rw, rw, ch, h, d0, d1, pad, pad}
// row stride 288 B == 72 banks == 8 (mod 64) -> bank group 8*((d+k) mod 8)

__device__ __forceinline__ float rqs_logp(float x,
    float b1, float b2, float b3, float b4, float b5, float b6, float b7,
    const char* __restrict__ dimbase) {
  float xc = fminf(fmaxf(x, -BOUND), BOUND);       // v_med3_f32
  // binary search over knots cumw[1..7] -> byte offset idx*32
  bool c4 = xc >= b4;
  float bm = c4 ? b6 : b2;
  float bl = c4 ? b5 : b1;
  float bh = c4 ? b7 : b3;
  bool c2 = xc >= bm;
  float bf = c2 ? bh : bl;
  bool c1 = xc >= bf;
  int off = (c4 ? 128 : 0) + (c2 ? 64 : 0) + (c1 ? 32 : 0);  // v_add3

  float4 a  = *(const float4*)(dimbase + off);        // ds_load_b128
  float2 dd = *(const float2*)(dimbase + off + 16);   // ds_load_b64 (offset imm)
  float rw = a.y, ch = a.z, h = a.w;
  float d0 = dd.x, d1 = dd.y;

  float delta = h * rw;
  float th   = fmaf(xc, rw, a.x);                  // (xc - cw) * rw
  float omt  = 1.0f - th;
  float th2  = th * th;
  float t1m  = th * omt;
  float omt2 = omt * omt;
  float s     = fmaf(-2.0f, delta, d0 + d1);
  float denom = fmaf(s, t1m, delta);
  float rden  = __builtin_amdgcn_rcpf(denom);      // v_rcp_f32 (1 ulp)
  float num = fmaf(delta, th2, d0 * t1m);
  float z   = fmaf(h * num, rden, ch);
  float u = delta * t1m;
  float A = fmaf(d1, th2, fmaf(d0, omt2, u + u));
  float q = delta * rden;
  float ld = __logf(q * q * A);                    // log(d^2 A) - 2 log(denom)
  bool inside = fabsf(x) <= BOUND;                 // NaN -> false -> identity
  z  = inside ? z  : x;
  ld = inside ? ld : 0.0f;
  return fmaf(-0.5f * z, z, ld - HALF_LOG_2PI);
}

__global__ void __launch_bounds__(256)
marginal_flow_bank_kernel(const float* __restrict__ x,
                          const float* __restrict__ uw,
                          const float* __restrict__ uh,
                          const float* __restrict__ ud,
                          float* __restrict__ out, int total4) {
  __shared__ float4 s_rec[DIMS * (ROWB / 16)];  // 9216 B padded records
  __shared__ float4 s_bnd[DIMS * 2];            // 1024 B: cumw[0..7] per dim

  // --- knot preprocessing: one lane per dim (wave 0), once per block ---
  if (threadIdx.x < DIMS) {
    const int d = threadIdx.x;
    float cumw[KBINS + 1], cumh[KBINS + 1], dv[KBINS + 1];
    {
      float u[KBINS]; float m = -1e30f;
#pragma unroll
      for (int k = 0; k < KBINS; ++k) { u[k] = uw[d * KBINS + k]; m = fmaxf(m, u[k]); }
      float s = 0.0f;
#pragma unroll
      for (int k = 0; k < KBINS; ++k) { u[k] = expf(u[k] - m); s += u[k]; }
      float inv = (1.0f - MINW * KBINS) / s;
      float c = 0.0f; cumw[0] = -BOUND;
#pragma unroll
      for (int k = 0; k < KBINS; ++k) {
        c += MINW + inv * u[k];
        cumw[k + 1] = 2.0f * BOUND * c - BOUND;
      }
      cumw[KBINS] = BOUND;
    }
    {
      float u[KBINS]; float m = -1e30f;
#pragma unroll
      for (int k = 0; k < KBINS; ++k) { u[k] = uh[d * KBINS + k]; m = fmaxf(m, u[k]); }
      float s = 0.0f;
#pragma unroll
      for (int k = 0; k < KBINS; ++k) { u[k] = expf(u[k] - m); s += u[k]; }
      float inv = (1.0f - MINH * KBINS) / s;
      float c = 0.0f; cumh[0] = -BOUND;
#pragma unroll
      for (int k = 0; k < KBINS; ++k) {
        c += MINH + inv * u[k];
        cumh[k + 1] = 2.0f * BOUND * c - BOUND;
      }
      cumh[KBINS] = BOUND;
    }
    dv[0] = 1.0f; dv[KBINS] = 1.0f;  // boundary pad: softplus(c) + MIND == 1
#pragma unroll
    for (int k = 1; k < KBINS; ++k) {
      float t = ud[d * (KBINS - 1) + k - 1];
      float sp = (t > 20.0f) ? t : log1pf(expf(t));
      dv[k] = MIND + sp;
    }
#pragma unroll
    for (int k = 0; k < KBINS; ++k) {
      float w  = cumw[k + 1] - cumw[k];
      float h  = cumh[k + 1] - cumh[k];
      float rw = 1.0f / w;
      s_rec[d * (ROWB / 16) + 2 * k]     = make_float4(-cumw[k] * rw, rw, cumh[k], h);
      s_rec[d * (ROWB / 16) + 2 * k + 1] = make_float4(dv[k], dv[k + 1], 0.0f, 0.0f);
    }
    s_bnd[2 * d]     = make_float4(cumw[0], cumw[1], cumw[2], cumw[3]);
    s_bnd[2 * d + 1] = make_float4(cumw[4], cumw[5], cumw[6], cumw[7]);
  }
  __syncthreads();

  const int gid    = blockIdx.x * blockDim.x + threadIdx.x;
  const int stride = gridDim.x * blockDim.x;   // blockDim=256 -> stride % 8 == 0
  const int dbase  = (gid & 7) * 4;            // this lane's 4 dims, loop-invariant

  // hoist per-dim knot boundaries + LDS row bases into VGPRs
  float bnd[4][8];
  const char* dbp[4];
#pragma unroll
  for (int j = 0; j < 4; ++j) {
    int d = dbase + j;
    float4 lo = s_bnd[2 * d], hi = s_bnd[2 * d + 1];
    bnd[j][1] = lo.y; bnd[j][2] = lo.z; bnd[j][3] = lo.w;
    bnd[j][4] = hi.x; bnd[j][5] = hi.y; bnd[j][6] = hi.z; bnd[j][7] = hi.w;
    dbp[j] = (const char*)s_rec + d * ROWB;
  }

  const float4* __restrict__ xin = (const float4*)x;
  float4* __restrict__ o = (float4*)out;
#pragma unroll 2
  for (int i = gid; i < total4; i += stride) {
    __builtin_prefetch(xin + i + stride, 0, 1);  // global_prefetch_b8 (speculative)
    float4 v = xin[i];
    float4 r;
    r.x = rqs_logp(v.x, bnd[0][1], bnd[0][2], bnd[0][3], bnd[0][4], bnd[0][5], bnd[0][6], bnd[0][7], dbp[0]);
    r.y = rqs_logp(v.y, bnd[1][1], bnd[1][2], bnd[1][3], bnd[1][4], bnd[1][5], bnd[1][6], bnd[1][7], dbp[1]);
    r.z = rqs_logp(v.z, bnd[2][1], bnd[2][2], bnd[2][3], bnd[2][4], bnd[2][5], bnd[2][6], bnd[2][7], dbp[2]);
    r.w = rqs_logp(v.w, bnd[3][1], bnd[3][2], bnd[3][3], bnd[3][4], bnd[3][5], bnd[3][6], bnd[3][7], dbp[3]);
    o[i] = r;
  }
}

extern "C" void kernel_launch(void* const* d_in, const int* in_sizes, int n_in,
                              void* d_out, int out_size, void* d_ws, size_t ws_size,
                              hipStream_t stream) {
  (void)n_in; (void)out_size; (void)d_ws; (void)ws_size;
  const float* x  = (const float*)d_in[0];
  const float* uw = (const float*)d_in[1];
  const float* uh = (const float*)d_in[2];
  const float* ud = (const float*)d_in[3];
  float* out = (float*)d_out;

  int total  = in_sizes[0];       // N*D = 33,554,432 (divisible by 4)
  int total4 = total >> 2;        // 8,388,608 float4s
  int threads = 256;
  int blocks  = 2048;             // 524,288 lanes -> 16 float4s per lane
  if ((total4 + threads - 1) / threads < blocks)
    blocks = (total4 + threads - 1) / threads;

  marginal_flow_bank_kernel<<<blocks, threads, 0, stream>>>(x, uw, uh, ud, out, total4);
}